// SenseContextAverage_50491635532436
// MI455X (gfx1250) — compile-verified
//
#include <hip/hip_runtime.h>
#include <math.h>

// ---------------------------------------------------------------------------
// Problem constants (from the reference)
// ---------------------------------------------------------------------------
#define T_DIM   32
#define B_DIM   32
#define D_DIM   512
#define G_DIM   32
#define S_DIM   40000
#define TB_DIM  (T_DIM * B_DIM)      // 1024 (t,b) pairs
#define INV_NUM_C (1.0f / 20.0f)
#define COS_EPS 1e-8f

typedef float v2f __attribute__((ext_vector_type(2)));
typedef float v4f __attribute__((ext_vector_type(4)));
typedef float v8f __attribute__((ext_vector_type(8)));

#define WAVES_PER_BLOCK 8

// ---------------------------------------------------------------------------
// Kernel 1: one wave per (t,b).
//  - build lc = (loc+emb)/20 in LDS (2KB per wave), wave-reduce ||lc||^2
//  - gather the 32 SC rows; per K-chunk of 4 feed V_WMMA_F32_16X16X4_F32
//    with A = lc chunk broadcast over rows, B = 16 gathered rows (x2 accums)
//  - accumulate per-row sum-of-squares alongside the loads
//  - cos = dot / (max(||lc||,eps) * max(||row||,eps)); shuffle-argmin over g
//  - lane 0 writes chosen sense id to scratch
// ---------------------------------------------------------------------------
__global__ __launch_bounds__(WAVES_PER_BLOCK * 32)
void sense_argmin_wmma_kernel(const float* __restrict__ emb,     // [T,B,D]
                              const float* __restrict__ locctx,  // [T,B,D]
                              const float* __restrict__ SC,      // [S,D]
                              const int*   __restrict__ nbr,     // [T,B,G]
                              int* __restrict__ chosen_out)      // [TB]
{
    __shared__ float lds_lc[WAVES_PER_BLOCK * D_DIM];

    const int lane = threadIdx.x & 31;
    const int wave = threadIdx.x >> 5;
    const int tb   = blockIdx.x * WAVES_PER_BLOCK + wave;   // grid = TB/8 exactly

    float* lc = &lds_lc[wave * D_DIM];

    // ---- lc into LDS + ||lc||^2 -------------------------------------------
    // emb/locctx are read exactly once: non-temporal loads keep them from
    // displacing the SC table in cache.
    const float* e = emb    + (size_t)tb * D_DIM;
    const float* c = locctx + (size_t)tb * D_DIM;
    float ss = 0.0f;
#pragma unroll
    for (int i = 0; i < D_DIM / 32; ++i) {
        const int d = lane * (D_DIM / 32) + i;
        const float v = (__builtin_nontemporal_load(c + d) +
                         __builtin_nontemporal_load(e + d)) * INV_NUM_C;
        lc[d] = v;
        ss = fmaf(v, v, ss);
    }
#pragma unroll
    for (int off = 16; off >= 1; off >>= 1)
        ss += __shfl_xor(ss, off, 32);
    const float lc_n = fmaxf(sqrtf(ss), COS_EPS);
    __syncthreads();

    // ---- gather setup ------------------------------------------------------
    // WMMA f32 16x16x4 per-lane layout: lane = n + 16*h handles column/row
    //   g = lane&15, K sub-offset koff = 2*h within each chunk of 4.
    const int g    = lane & 15;
    const int koff = (lane >> 4) << 1;
    const int s1 = nbr[tb * G_DIM + g];          // rows for accumulator 1 (g 0..15)
    const int s2 = nbr[tb * G_DIM + g + 16];     // rows for accumulator 2 (g 16..31)
    const float* row1 = SC + (size_t)s1 * D_DIM + koff;
    const float* row2 = SC + (size_t)s2 * D_DIM + koff;
    const float* lcp  = lc + koff;

    v8f acc1 = {0.f, 0.f, 0.f, 0.f, 0.f, 0.f, 0.f, 0.f};
    v8f acc2 = {0.f, 0.f, 0.f, 0.f, 0.f, 0.f, 0.f, 0.f};
    float sq1 = 0.0f, sq2 = 0.0f;

#pragma unroll 4
    for (int k0 = 0; k0 < D_DIM; k0 += 4) {
        const v2f a  = *(const v2f*)(lcp  + k0);   // ds_load (pairs -> 2addr_b64)
        const v2f b1 = *(const v2f*)(row1 + k0);   // global_load_b64 (gather)
        const v2f b2 = *(const v2f*)(row2 + k0);
        sq1 = fmaf(b1.x, b1.x, sq1); sq1 = fmaf(b1.y, b1.y, sq1);
        sq2 = fmaf(b2.x, b2.x, sq2); sq2 = fmaf(b2.y, b2.y, sq2);
        // D = A(16x4, lc bcast) x B(4x16, gathered rows) + C
        acc1 = __builtin_amdgcn_wmma_f32_16x16x4_f32(
                   false, a, false, b1, (short)0, acc1, false, false);
        acc2 = __builtin_amdgcn_wmma_f32_16x16x4_f32(
                   false, a, false, b2, (short)0, acc2, false, false);
    }

    // ---- row norms: lanes ln and ln^16 hold complementary K halves --------
    sq1 += __shfl_xor(sq1, 16, 32);
    sq2 += __shfl_xor(sq2, 16, 32);
    const float n1 = fmaxf(sqrtf(sq1), COS_EPS);
    const float n2 = fmaxf(sqrtf(sq2), COS_EPS);

    // every lane holds dots[lane&15] (acc1) and dots[16+(lane&15)] (acc2)
    const float cos1 = acc1[0] / (lc_n * n1);
    const float cos2 = acc2[0] / (lc_n * n2);

    float bc; int bg;
    if (cos2 < cos1) { bc = cos2; bg = g + 16; }
    else             { bc = cos1; bg = g;      }

    // wave argmin (first-occurrence tie-break: smaller g wins on equality).
    // Lanes 16..31 hold exact duplicates of lanes 0..15, so reducing within
    // a 16-lane half (offsets 1,2,4,8) is sufficient.
#pragma unroll
    for (int off = 1; off <= 8; off <<= 1) {
        const float oc = __shfl_xor(bc, off, 32);
        const int   og = __shfl_xor(bg, off, 32);
        if (oc < bc || (oc == bc && og < bg)) { bc = oc; bg = og; }
    }

    if (lane == 0)
        chosen_out[tb] = nbr[tb * G_DIM + bg];
}

// ---------------------------------------------------------------------------
// Kernel 2: stream the [TB, S] log-prob matrix (164 MB) with 16B stores.
// value = log(1e-8) everywhere except column == chosen[row].
// The output is never re-read: non-temporal stores (TH_STORE_NT) keep this
// 164 MB stream from sweeping the 192 MB L2.
// ---------------------------------------------------------------------------
__global__ __launch_bounds__(256)
void fill_logprobs_kernel(const int* __restrict__ chosen,
                          float* __restrict__ out,
                          float log_eps, float log_ch)
{
    const int per_row4 = S_DIM / 4;                       // 10000 v4f per row
    const int i = blockIdx.x * 256 + threadIdx.x;         // v4f index
    if (i >= TB_DIM * per_row4) return;

    const int row = i / per_row4;
    const int c4  = i - row * per_row4;
    const int rel = chosen[row] - (c4 << 2);              // chosen col - base col

    v4f v;
    v[0] = (rel == 0) ? log_ch : log_eps;
    v[1] = (rel == 1) ? log_ch : log_eps;
    v[2] = (rel == 2) ? log_ch : log_eps;
    v[3] = (rel == 3) ? log_ch : log_eps;
    __builtin_nontemporal_store(v, (v4f*)out + i);        // global_store_b128 NT
}

// ---------------------------------------------------------------------------
extern "C" void kernel_launch(void* const* d_in, const int* in_sizes, int n_in,
                              void* d_out, int out_size, void* d_ws, size_t ws_size,
                              hipStream_t stream)
{
    (void)in_sizes; (void)n_in; (void)out_size; (void)ws_size;

    const float* emb = (const float*)d_in[0];   // word_embeddings [T,B,D]
    const float* loc = (const float*)d_in[1];   // location_context [T,B,D]
    const float* SC  = (const float*)d_in[2];   // SC [S,D]
    const int*   nbr = (const int*)d_in[3];     // all_sense_neighbours [T,B,G]
    float* out = (float*)d_out;
    int* chosen = (int*)d_ws;                   // TB ints of scratch

    sense_argmin_wmma_kernel<<<dim3(TB_DIM / WAVES_PER_BLOCK),
                               dim3(WAVES_PER_BLOCK * 32), 0, stream>>>(
        emb, loc, SC, nbr, chosen);

    const float log_eps = logf(1e-8f);
    const float log_ch  = logf(1.0f - 1e-8f * (float)(S_DIM - 1));
    const int total4 = TB_DIM * (S_DIM / 4);
    fill_logprobs_kernel<<<dim3((total4 + 255) / 256), dim3(256), 0, stream>>>(
        chosen, out, log_eps, log_ch);
}